// Adjacent1d_8847632630018
// MI455X (gfx1250) — compile-verified
//
#include <hip/hip_runtime.h>

typedef __attribute__((ext_vector_type(2))) float v2f;
typedef __attribute__((ext_vector_type(8))) float v8f;

#define B_      4
#define IN_C    64
#define IN_S    16384
#define OUT_C   64
#define OUT_S   4096
#define CTOT    (IN_C * 4)        // 256 contraction length (i,k)

#define S_BLK   32                // s values per workgroup (coalesced weight rows)
#define O_TILE  16                // o values per workgroup (WMMA M)
#define CC      8                 // c-chunk staged in LDS per stage
#define O_STR   36                // padded o stride (floats): 16B-aligned for b128,
                                  // o*36 mod 64 is a permutation -> conflict-free half-wave
#define C_STR   (O_TILE * O_STR)  // 576 floats per c row (16B aligned)
#define NSTAGE  (CTOT / CC)       // 32

// gfx1250 async global->LDS path (ASYNCcnt-tracked), with safe fallback.
#if defined(__has_builtin)
#if __has_builtin(__builtin_amdgcn_global_load_async_to_lds_b128) && \
    __has_builtin(__builtin_amdgcn_s_wait_asynccnt)
#define HAVE_ASYNC_LDS 1
#endif
#endif
#ifndef HAVE_ASYNC_LDS
#define HAVE_ASYNC_LDS 0
#endif

// Exact param types per hipcc diagnostic:
//   arg0: int __attribute__((vector_size(16))) __device__ *   (addrspace 1)
//   arg1: matching vector pointer in LDS                      (addrspace 3)
typedef int v4i __attribute__((vector_size(16)));
typedef __attribute__((address_space(1))) v4i gv4i;
typedef __attribute__((address_space(3))) v4i lv4i;

__global__ __launch_bounds__(256)
void adj1d_wmma_kernel(const float* __restrict__ x,
                       const float* __restrict__ w,
                       const float* __restrict__ bias,
                       float* __restrict__ out)
{
    __shared__ float lds[CC * C_STR];   // 18 KB

    const int tid  = threadIdx.x;
    const int lane = tid & 31;
    const int wave = tid >> 5;          // 0..7, each wave owns 4 s values

    const int o_base = blockIdx.y * O_TILE;
    const int s_base = blockIdx.x * S_BLK;

    const int n    = lane & 15;         // D column = batch index (only 0..3 used)
    const int half = lane >> 4;         // lane half
    const int koff = half * 2;          // K offset this half holds (A & B operands)
    const int o_l  = lane & 15;         // A row = o within tile

    v8f acc[4] = {};

    for (int stage = 0; stage < NSTAGE; ++stage) {
        const int cb = stage * CC;

        __syncthreads();                // previous-stage readers done
#if HAVE_ASYNC_LDS
        // 4 x b128 async DMA per thread: 8 threads cover one (c,o) row of 32 s
        #pragma unroll
        for (int it = 0; it < 4; ++it) {
            const int q   = tid + it * 256;   // 0..1023 quads
            const int row = q >> 3;           // 0..127 = c*16 + o
            const int ssq = (q & 7) * 4;      // s quad within row
            const int c_l = row >> 4;
            const int o_r = row & 15;
            const float* gp = w + ((size_t)(cb + c_l) * OUT_C + (o_base + o_r)) * OUT_S
                                + (size_t)(s_base + ssq);
            float* lp = &lds[c_l * C_STR + o_r * O_STR + ssq];
            __builtin_amdgcn_global_load_async_to_lds_b128(
                (gv4i*)gp, (lv4i*)lp, 0, 0);
        }
        __builtin_amdgcn_s_wait_asynccnt(0);
#else
        // fallback: global->VGPR->LDS, 32-thread rows of 32 consecutive s
        {
            const int ss = tid & 31;
            const int r0 = tid >> 5;
            #pragma unroll
            for (int it = 0; it < (CC * O_TILE) / 8; ++it) {
                const int row = r0 + it * 8;   // 0..127
                const int c_l = row >> 4;
                const int o_r = row & 15;
                lds[c_l * C_STR + o_r * O_STR + ss] =
                    w[((size_t)(cb + c_l) * OUT_C + (o_base + o_r)) * OUT_S
                      + (size_t)(s_base + ss)];
            }
        }
#endif
        __syncthreads();                // stage data visible to all waves

        #pragma unroll
        for (int hh = 0; hh < 2; ++hh) {       // two K=4 steps per chunk
            const int i_ch = (cb >> 2) + hh;   // input channel (c0 % 4 == 0)
            const int cl0  = hh * 4 + koff;    // LDS c index for this lane's K pair
            #pragma unroll
            for (int j = 0; j < 4; ++j) {
                const int s_loc = wave * 4 + j;
                const int s     = s_base + s_loc;

                v2f a;
                a.x = lds[(cl0    ) * C_STR + o_l * O_STR + s_loc];
                a.y = lds[(cl0 + 1) * C_STR + o_l * O_STR + s_loc];

                v2f b; b.x = 0.0f; b.y = 0.0f;
                if (n < B_) {
                    const float* xp = x + ((size_t)n * IN_C + i_ch) * IN_S
                                        + (size_t)(4 * s + koff);
                    b.x = xp[0];               // x[b, i, 4s+koff]
                    b.y = xp[1];               // x[b, i, 4s+koff+1]
                }

                acc[j] = __builtin_amdgcn_wmma_f32_16x16x4_f32(
                    false, a, false, b, (short)0, acc[j], false, false);
            }
        }
    }

    // Epilogue: D[m,n] layout -> out[b=n, o_base+m, s], add bias
    #pragma unroll
    for (int j = 0; j < 4; ++j) {
        const int s = s_base + wave * 4 + j;
        if (n < B_) {
            #pragma unroll
            for (int r = 0; r < 8; ++r) {
                const int m = r + 8 * half;
                const int o = o_base + m;
                out[((size_t)n * OUT_C + o) * OUT_S + s] =
                    acc[j][r] + bias[(size_t)o * OUT_S + s];
            }
        }
    }
}

extern "C" void kernel_launch(void* const* d_in, const int* in_sizes, int n_in,
                              void* d_out, int out_size, void* d_ws, size_t ws_size,
                              hipStream_t stream) {
    (void)in_sizes; (void)n_in; (void)out_size; (void)d_ws; (void)ws_size;
    const float* x    = (const float*)d_in[0];
    const float* w    = (const float*)d_in[1];
    const float* bias = (const float*)d_in[2];
    float* out        = (float*)d_out;

    dim3 grid(OUT_S / S_BLK, OUT_C / O_TILE);   // 128 x 4 workgroups
    adj1d_wmma_kernel<<<grid, 256, 0, stream>>>(x, w, bias, out);
}